// ThermoPT_22797686407686
// MI455X (gfx1250) — compile-verified
//
#include <hip/hip_runtime.h>
#include <stdint.h>

// Problem constants (from reference)
#define BB 16
#define HH 120
#define WW 160
#define DD 401
#define WTILE 16              // w-columns per block
#define NTILES (WW / WTILE)   // 10
#define NTHREADS 256          // 8 waves (wave32)
#define TILE_ELEMS (WTILE * DD)   // 6416 floats, contiguous in LDS and global

// Gaussian constants: sigma+eps = 0.50000001
#define SIG_EPS   0.50000001f
#define INV_SIG   (1.0f / SIG_EPS)                       // ~1.99999996
#define GNORM     (INV_SIG * 0.3989422804014327f)        // 1/((sig+eps)*sqrt(2pi))

typedef unsigned int u32x4 __attribute__((ext_vector_type(4)));
typedef int          s32x4 __attribute__((ext_vector_type(4)));
typedef int          s32x8 __attribute__((ext_vector_type(8)));

// Pack a D# (groups 0+1) and issue a TDM op. dims: 2-D tile of 4-byte elems.
__device__ __forceinline__ void tdm_issue(bool is_store,
                                          unsigned lds_off,
                                          unsigned long long ga,
                                          unsigned tensor_d0, unsigned tensor_d1,
                                          unsigned tile_d0, unsigned tile_d1,
                                          unsigned d0_stride)
{
    // D# group 0 (ISA 8.3): count=1 | lds_addr | global_addr[56:0] | type=2
    u32x4 g0;
    g0[0] = 1u;                                   // count=1, user mode
    g0[1] = lds_off;                              // lds_addr (bytes)
    g0[2] = (unsigned)ga;                         // global_addr[31:0]
    g0[3] = (unsigned)(ga >> 32) | (2u << 30);    // global_addr[56:32] | type=2

    // D# group 1 (ISA 8.4): data_size=2 (4B), 2-D tile
    s32x8 g1;
    g1[0] = (int)(2u << 16);                          // mask=0, data_size=4B
    g1[1] = (int)(tensor_d0 << 16);                   // tensor_dim0[15:0]
    g1[2] = (int)((tensor_d0 >> 16) | (tensor_d1 << 16)); // dim0 hi | dim1 lo
    g1[3] = (int)((tensor_d1 >> 16) | (tile_d0 << 16));   // dim1 hi | tile_dim0
    g1[4] = (int)tile_d1;                             // tile_dim1 | tile_dim2=0
    g1[5] = (int)d0_stride;                           // tensor_dim0_stride[31:0]
    g1[6] = 0;
    g1[7] = 0;

    s32x4 gz = {0, 0, 0, 0};                          // groups 2/3 unused (2-D)

#if defined(__clang_major__) && (__clang_major__ >= 23)
    s32x8 gz8 = {0, 0, 0, 0, 0, 0, 0, 0};
    if (is_store) __builtin_amdgcn_tensor_store_from_lds(g0, g1, gz, gz, gz8, 0);
    else          __builtin_amdgcn_tensor_load_to_lds  (g0, g1, gz, gz, gz8, 0);
#else
    if (is_store) __builtin_amdgcn_tensor_store_from_lds(g0, g1, gz, gz, 0);
    else          __builtin_amdgcn_tensor_load_to_lds  (g0, g1, gz, gz, 0);
#endif
}

__global__ __launch_bounds__(NTHREADS)
void ThermoPT_dim_bev_kernel(const float* __restrict__ depth,
                             float* __restrict__ out)
{
    // LDS: depth tile (row h of 16 contiguous w's) + per-tile histogram
    __shared__ float s_depth[HH * WTILE];      // 7.5 KB
    __shared__ float s_hist[TILE_ELEMS];       // 25.7 KB

    const int tid = threadIdx.x;
    const int wx  = blockIdx.x;                // w-tile index, 0..9
    const int b   = blockIdx.y;                // batch,        0..15
    const int w0  = wx * WTILE;

    // ---- Phase 1: wave 0 issues TDM DMA of depth[b, :, w0:w0+16] -> LDS ----
    if (tid < 32) {   // wave-uniform branch: only wave 0 reaches the TDM op
        unsigned long long ga_in =
            (unsigned long long)(const void*)(depth + (size_t)b * (HH * WW) + w0);
        tdm_issue(/*store=*/false,
                  (unsigned)(uintptr_t)(&s_depth[0]), ga_in,
                  /*tensor*/ WW, HH, /*tile*/ WTILE, HH, /*stride*/ WW);
    }

    // ---- Phase 2 (overlapped with DMA): zero the LDS histogram ----
    #pragma unroll 4
    for (int i = tid; i < TILE_ELEMS; i += NTHREADS)
        s_hist[i] = 0.0f;

    if (tid < 32)
        __builtin_amdgcn_s_wait_tensorcnt(0);   // wave 0: input DMA complete
    __syncthreads();                            // publish tile + zeroed hist

    // ---- Phase 3: sparse Gaussian scatter into LDS histogram ----
    // fp32 fact: bins are 20 apart, sigma=0.5 => only the nearest bin
    // (|20k - d| < ~7.2) is representable; all other bins underflow to
    // exactly 0, matching the dense reference elementwise.
    #pragma unroll 2
    for (int i = tid; i < HH * WTILE; i += NTHREADS) {
        float dv = s_depth[i];
        // valid: finite, > 1e-6, <= 8000 (NaN/inf fail these compares)
        if (dv > 1e-6f && dv <= 8000.0f) {
            int wi = i & (WTILE - 1);
            int k  = __float2int_rn(dv * 0.05f);        // nearest bin, v_rndne
            float z   = ((float)k * 20.0f - dv) * INV_SIG;
            float val = __expf(-0.5f * z * z) * GNORM;  // v_exp_f32
            atomicAdd(&s_hist[wi * DD + k], val);       // ds_add_f32
        }
    }
    __syncthreads();   // all scatter adds visible before DMA-out reads LDS

    // ---- Phase 4: TDM DMA of the contiguous result tile LDS -> global ----
    // out[b, w0+wi, d] is contiguous over (wi,d): one 1-D 25.7KB descriptor.
    if (tid < 32) {
        unsigned long long ga_out =
            (unsigned long long)(void*)(out + (size_t)(b * WW + w0) * DD);
        tdm_issue(/*store=*/true,
                  (unsigned)(uintptr_t)(&s_hist[0]), ga_out,
                  /*tensor*/ TILE_ELEMS, 1, /*tile*/ TILE_ELEMS, 1,
                  /*stride*/ TILE_ELEMS);
        __builtin_amdgcn_s_wait_tensorcnt(0);   // drain before wave retires
    }
    // waves 1..7 exit immediately; wave 0 holds the workgroup (and its LDS)
    // alive until the store DMA completes (S_ENDPGM also implies wait-idle).
}

extern "C" void kernel_launch(void* const* d_in, const int* in_sizes, int n_in,
                              void* d_out, int out_size, void* d_ws, size_t ws_size,
                              hipStream_t stream) {
    (void)in_sizes; (void)n_in; (void)out_size; (void)d_ws; (void)ws_size;
    const float* depth = (const float*)d_in[0];
    float* out = (float*)d_out;
    dim3 grid(NTILES, BB);     // 10 x 16 = 160 blocks
    dim3 block(NTHREADS);      // 8 wave32 waves
    ThermoPT_dim_bev_kernel<<<grid, block, 0, stream>>>(depth, out);
}